// HierarchicalSoftmax_4544075399420
// MI455X (gfx1250) — compile-verified
//
#include <hip/hip_runtime.h>
#include <cstdint>
#include <cstddef>

// CDNA5 f32 WMMA operand types: 16x16x4 f32 -> A,B are 2 VGPRs (float2), C/D 8 VGPRs.
typedef float v2f __attribute__((ext_vector_type(2)));
typedef float v8f __attribute__((ext_vector_type(8)));

// One wave (32 lanes) per (16-sample tile, level-chunk).
//   scores[m] for the tile = diag( Emb_tile(16xD) * W_gathered(Dx16) )
// computed as a K-loop of V_WMMA_F32_16X16X4_F32 accumulations.
//
// Lane mapping (from CDNA5 ISA 7.12.2, 32-bit operands):
//   A (16x4, MxK): lane L -> m = L%16 ; VGPR0/1 hold k = 2*(L/16), 2*(L/16)+1
//   B (4x16, KxN): lane L -> n = L%16 ; VGPR0/1 hold k = 2*(L/16), 2*(L/16)+1  (mirror)
//   C/D (16x16):   VGPR j: lanes 0-15 -> M=j,  N=lane ; lanes 16-31 -> M=j+8, N=lane-16
//   => diagonal element for sample m: lane m, comp m   (m < 8)
//                                     lane m+16, comp m-8 (m >= 8)
// The owner lane of sample m satisfies (ownerLane % 16) == m, so the same lane that
// loads node/dir/bias for sample n==m also owns its score and running product.
__global__ __launch_bounds__(32)
void hs_wmma_chunk_kernel(const float* __restrict__ emb,    // [B, D]
                          const float* __restrict__ W,      // [V-1, D]
                          const float* __restrict__ bias,   // [V-1]
                          const int*   __restrict__ target, // [B]
                          const int*   __restrict__ pnodes, // [V, L]
                          const int*   __restrict__ pdirs,  // [V, L]
                          float* __restrict__ dst,          // [nchunks, B] (or [B] if 1 chunk)
                          int B, int D, int L, int CH)
{
    const int lane = threadIdx.x;        // 0..31, wave32
    const int n    = lane & 15;          // sample slot within tile (N column / A row)
    const int half = lane >> 4;          // K sub-offset selector
    const int b0   = blockIdx.x * 16;
    int bn = b0 + n;
    if (bn >= B) bn = B - 1;             // clamp; keeps EXEC all-1s for WMMA

    const int chunk = blockIdx.y;
    const int lbeg  = chunk * CH;
    const int lend  = (lbeg + CH < L) ? (lbeg + CH) : L;

    const long t = (long)target[bn];
    const int* __restrict__ prow = pnodes + t * (long)L;
    const int* __restrict__ drow = pdirs  + t * (long)L;

    // Path validity prefix: node==0 only at root (last valid entry) or padding,
    // so level l is on-path iff all earlier nodes are nonzero.
    bool valid = true;
    for (int l = 0; l < lbeg; ++l) valid = valid && (prow[l] != 0);

    // This lane's A source: emb row of its sample, shifted by its K sub-offset.
    const float* __restrict__ Arow = emb + (size_t)bn * (size_t)D + 2 * half;

    const bool isOwner = (lane < 8) || (lane >= 24);
    const int  oc      = (lane < 8) ? lane : (lane - 24);   // diag component index

    float prod = 1.0f;
    const int kq = D >> 2;               // number of 16x16x4 steps (D=512 -> 128)

    for (int l = lbeg; l < lend; ++l) {
        const int node = prow[l];
        const int dir  = drow[l];
        const float* __restrict__ Brow = W + (size_t)node * (size_t)D + 2 * half;

        // Two accumulators break the WMMA->WMMA RAW chain on C.
        v8f c0 = {0.f,0.f,0.f,0.f,0.f,0.f,0.f,0.f};
        v8f c1 = {0.f,0.f,0.f,0.f,0.f,0.f,0.f,0.f};
        int kk = 0;
#pragma unroll 4
        for (; kk + 1 < kq; kk += 2) {
            v2f a0 = *(const v2f*)(Arow + 4 * kk);
            v2f bb0 = *(const v2f*)(Brow + 4 * kk);
            c0 = __builtin_amdgcn_wmma_f32_16x16x4_f32(false, a0, false, bb0,
                                                       (short)0, c0, false, false);
            v2f a1 = *(const v2f*)(Arow + 4 * kk + 4);
            v2f bb1 = *(const v2f*)(Brow + 4 * kk + 4);
            c1 = __builtin_amdgcn_wmma_f32_16x16x4_f32(false, a1, false, bb1,
                                                       (short)0, c1, false, false);
        }
        for (; kk < kq; ++kk) {          // tail (D % 8 != 0)
            v2f a0 = *(const v2f*)(Arow + 4 * kk);
            v2f bb0 = *(const v2f*)(Brow + 4 * kk);
            c0 = __builtin_amdgcn_wmma_f32_16x16x4_f32(false, a0, false, bb0,
                                                       (short)0, c0, false, false);
        }
        v8f cs = c0 + c1;

        // Branch-free epilogue (keeps EXEC untouched inside the level loop).
        float sc = (oc == 0) ? cs.s0 : (oc == 1) ? cs.s1 : (oc == 2) ? cs.s2 :
                   (oc == 3) ? cs.s3 : (oc == 4) ? cs.s4 : (oc == 5) ? cs.s5 :
                   (oc == 6) ? cs.s6 : cs.s7;
        sc += bias[node];
        const float sgn = dir ? sc : -sc;          // dir==1 -> sigmoid(s), else sigmoid(-s)
        float f = 1.0f / (1.0f + expf(-sgn));
        f = valid ? f : 1.0f;
        prod *= f;
        valid = valid && (node != 0);              // root (node 0) ends the path
    }

    if (isOwner && (b0 + n) < B)
        dst[(size_t)chunk * (size_t)B + (size_t)(b0 + n)] = prod;
}

// Combine per-chunk partial products: out[b] = prod_c ws[c][b].
__global__ void hs_product_kernel(const float* __restrict__ ws,
                                  float* __restrict__ out, int B, int nchunks)
{
    const int b = blockIdx.x * blockDim.x + threadIdx.x;
    if (b >= B) return;
    float p = 1.0f;
    for (int c = 0; c < nchunks; ++c) p *= ws[(size_t)c * (size_t)B + b];
    out[b] = p;
}

extern "C" void kernel_launch(void* const* d_in, const int* in_sizes, int n_in,
                              void* d_out, int out_size, void* d_ws, size_t ws_size,
                              hipStream_t stream)
{
    const float* emb    = (const float*)d_in[0];   // embeddings [B,D] f32
    const float* W      = (const float*)d_in[1];   // W [V-1,D] f32
    const float* bias   = (const float*)d_in[2];   // b [V-1] f32
    const int*   target = (const int*)d_in[3];     // target [B] i32
    const int*   pnodes = (const int*)d_in[4];     // path_nodes [V,L] i32
    const int*   pdirs  = (const int*)d_in[5];     // path_dirs  [V,L] i32
    // d_in[6] = path_mask: intentionally unused (mask derived from node==0 sentinel,
    // avoiding any assumption about how bool arrays are marshalled).

    const int B   = in_sizes[3];
    const int D   = in_sizes[0] / B;
    const int Vm1 = in_sizes[1] / D;
    const int V   = Vm1 + 1;
    const int L   = in_sizes[4] / V;

    float* out = (float*)d_out;

    // Pick the smallest level-chunk size whose workspace fits; fall back to a
    // single serial chunk (writes d_out directly, no reduce kernel) if needed.
    int CH = 1;
    while (CH < L && (size_t)((L + CH - 1) / CH) * (size_t)B * sizeof(float) > ws_size)
        ++CH;
    int nchunks = (L + CH - 1) / CH;
    if ((size_t)nchunks * (size_t)B * sizeof(float) > ws_size) { CH = L; nchunks = 1; }

    float* dst = (nchunks > 1) ? (float*)d_ws : out;

    dim3 grid((B + 15) / 16, nchunks);
    hs_wmma_chunk_kernel<<<grid, 32, 0, stream>>>(emb, W, bias, target, pnodes, pdirs,
                                                  dst, B, D, L, CH);
    if (nchunks > 1) {
        hs_product_kernel<<<(B + 255) / 256, 256, 0, stream>>>((const float*)d_ws,
                                                               out, B, nchunks);
    }
}